// Q_Act_u_13176959664396
// MI455X (gfx1250) — compile-verified
//
#include <hip/hip_runtime.h>
#include <math.h>

// Problem constants (from reference): x (8, 4096, 2048) fp32, s (4096,)
#define NB    8
#define NL    4096
#define ND    2048
#define NREF  10
#define NG    410              // ceil((4096 + (-4096 % 10)) / 10) = 4100/10
#define NROWS (NB * NG)        // 3280 (b,g) rows per step
#define TPB   256
#define EPT   8                // ND / TPB elements per thread
#define NWAVES (TPB / 32)      // 8 waves per block (wave32)

// Workspace layout (d_ws):
//   [0,64)            : unsigned flags[16]  (per-step global 'cond')
//   [64, 64+16384)    : float thr[4096]     (per-row threshold, overwritten per step)
//   [16448, +840KB)   : unsigned char obits[NROWS*TPB] (carried outlier bitmask)
#define WS_THR_OFF   64
#define WS_OBIT_OFF  (64 + 16384)

#if defined(__HIP_DEVICE_COMPILE__)
#if __has_builtin(__builtin_amdgcn_global_load_async_to_lds_b128)
#define USE_ASYNC_LDS 1
#endif
#endif

// Pointer types expected by the async-to-LDS builtin (from hipcc diagnostic):
// (AS1 int4*, AS3 int4*, imm offset, imm cpol)
typedef int vint4_t __attribute__((vector_size(16)));
typedef __attribute__((address_space(1))) vint4_t* gbl_v4i_ptr;
typedef __attribute__((address_space(3))) vint4_t* lds_v4i_ptr;

// ---- wave32 butterfly reductions (all lanes end with the result) ----
__device__ __forceinline__ float wave_sum(float v) {
  #pragma unroll
  for (int m = 16; m > 0; m >>= 1) v += __shfl_xor(v, m, 32);
  return v;
}
__device__ __forceinline__ float wave_max(float v) {
  #pragma unroll
  for (int m = 16; m > 0; m >>= 1) v = fmaxf(v, __shfl_xor(v, m, 32));
  return v;
}

// Block-wide reductions: per-wave shuffle, 8-value LDS combine. Values are
// non-negative (sums of squares / abs), so 0 is a valid identity for max.
__device__ __forceinline__ float block_sum(float v, float* red) {
  const int lane = threadIdx.x & 31, wid = threadIdx.x >> 5;
  v = wave_sum(v);
  if (lane == 0) red[wid] = v;
  __syncthreads();
  if (wid == 0) {
    float x = (lane < NWAVES) ? red[lane] : 0.0f;
    #pragma unroll
    for (int m = NWAVES / 2; m > 0; m >>= 1) x += __shfl_xor(x, m, 32);
    if (lane == 0) red[0] = x;
  }
  __syncthreads();
  float r = red[0];
  __syncthreads();
  return r;
}
__device__ __forceinline__ float block_max(float v, float* red) {
  const int lane = threadIdx.x & 31, wid = threadIdx.x >> 5;
  v = wave_max(v);
  if (lane == 0) red[wid] = v;
  __syncthreads();
  if (wid == 0) {
    float x = (lane < NWAVES) ? red[lane] : 0.0f;
    #pragma unroll
    for (int m = NWAVES / 2; m > 0; m >>= 1) x = fmaxf(x, __shfl_xor(x, m, 32));
    if (lane == 0) red[0] = x;
  }
  __syncthreads();
  float r = red[0];
  __syncthreads();
  return r;
}

__global__ void init_flags_kernel(unsigned* flags) {
  if (threadIdx.x < 16) flags[threadIdx.x] = 0u;
}

// Pass A (per step): per-row stats with previous mask; sets global cond flag.
template <bool FIRST>
__global__ void stats_kernel(const float* __restrict__ x,
                             const float* __restrict__ s,
                             unsigned* __restrict__ flags,
                             float* __restrict__ thrbuf,
                             const unsigned char* __restrict__ obits,
                             int step) {
  __shared__ float rowbuf[ND];
  __shared__ float red[16];

  const int r = blockIdx.x;
  const int g = r % NG;
  const int b = r / NG;
  const int l = g * NREF + step;
  if (l >= NL) return;  // padded rows: provably inert, skip (block-uniform)

  const int t = threadIdx.x;
  const float* gp = x + ((size_t)b * NL + (size_t)l) * ND + (size_t)t * EPT;
  float* lp = &rowbuf[t * EPT];

#if defined(USE_ASYNC_LDS)
  // CDNA5 async global->LDS DMA path; tracked by ASYNCcnt.
  __builtin_amdgcn_global_load_async_to_lds_b128(
      (gbl_v4i_ptr)gp, (lds_v4i_ptr)lp, 0, 0);
  __builtin_amdgcn_global_load_async_to_lds_b128(
      (gbl_v4i_ptr)(gp + 4), (lds_v4i_ptr)(lp + 4), 0, 0);
#if __has_builtin(__builtin_amdgcn_s_wait_asynccnt)
  __builtin_amdgcn_s_wait_asynccnt(0);
#else
  asm volatile("s_wait_asynccnt 0" ::: "memory");
#endif
#else
  {
    float4 v0 = *(const float4*)gp;
    float4 v1 = *(const float4*)(gp + 4);
    *(float4*)lp       = v0;
    *(float4*)(lp + 4) = v1;
  }
#endif
  __syncthreads();

  const unsigned ob = FIRST ? 0u : (unsigned)obits[(size_t)r * TPB + t];

  float a[EPT];
  float ps = 0.0f, pm = 0.0f;
  #pragma unroll
  for (int j = 0; j < EPT; ++j) {
    const float xv = rowbuf[t * EPT + j];
    const float av = ((ob >> j) & 1u) ? 0.0f : fabsf(xv);
    a[j] = av;
    ps += av;
    pm = fmaxf(pm, av);
  }

  const float sum  = block_sum(ps, red);
  const float amax = block_max(pm, red);
  const float mean = sum / (float)ND;

  float pv = 0.0f;
  #pragma unroll
  for (int j = 0; j < EPT; ++j) {
    const float d = a[j] - mean;
    pv += d * d;
  }
  const float var = block_sum(pv, red);
  const float thr = mean + 3.0f * sqrtf(var / (float)(ND - 1));

  if (t == 0) {
    thrbuf[r] = thr;
    const float dyn_s = amax / 7.0f;  // QMAX = 7
    if (dyn_s > s[l]) atomicOr(&flags[step], 1u);
  }
}

// Pass B (per step): consume global cond, update mask, re-quantize, write out.
template <bool FIRST>
__global__ void quant_kernel(const float* __restrict__ x,
                             const float* __restrict__ s,
                             const unsigned* __restrict__ flags,
                             const float* __restrict__ thrbuf,
                             unsigned char* __restrict__ obits,
                             float* __restrict__ out,
                             int step) {
  __shared__ float red[16];

  const int r = blockIdx.x;
  const int g = r % NG;
  const int b = r / NG;
  const int l = g * NREF + step;
  if (l >= NL) return;

  const int t = threadIdx.x;
  const size_t base = ((size_t)b * NL + (size_t)l) * ND + (size_t)t * EPT;

  const float4 v0 = *(const float4*)(x + base);
  const float4 v1 = *(const float4*)(x + base + 4);
  float xv[EPT] = {v0.x, v0.y, v0.z, v0.w, v1.x, v1.y, v1.z, v1.w};

  unsigned ob = FIRST ? 0u : (unsigned)obits[(size_t)r * TPB + t];
  const bool cond = (flags[step] != 0u);
  const float thr = thrbuf[r];

  if (cond) {
    #pragma unroll
    for (int j = 0; j < EPT; ++j) {
      const float av = ((ob >> j) & 1u) ? 0.0f : fabsf(xv[j]);
      if (av > thr) ob |= (1u << j);
    }
  }

  float pm = 0.0f;
  #pragma unroll
  for (int j = 0; j < EPT; ++j)
    if ((ob >> j) & 1u) pm = fmaxf(pm, fabsf(xv[j]));
  const float omax = block_max(pm, red);
  const float osc  = fmaxf(omax, 1e-8f) / 127.0f;  // O_QMAX = 127
  const float sc   = s[l];

  float q[EPT];
  #pragma unroll
  for (int j = 0; j < EPT; ++j) {
    if ((ob >> j) & 1u) {
      float rq = rintf(xv[j] / osc);               // jnp.round = half-to-even
      rq = fminf(fmaxf(rq, -127.0f), 127.0f);
      q[j] = rq * osc;
    } else {
      float rq = rintf(xv[j] / sc);
      rq = fminf(fmaxf(rq, -7.0f), 7.0f);
      q[j] = rq * sc;
    }
  }

  float4 o0 = {q[0], q[1], q[2], q[3]};
  float4 o1 = {q[4], q[5], q[6], q[7]};
  *(float4*)(out + base)     = o0;
  *(float4*)(out + base + 4) = o1;

  if (FIRST || cond) obits[(size_t)r * TPB + t] = (unsigned char)ob;
}

extern "C" void kernel_launch(void* const* d_in, const int* in_sizes, int n_in,
                              void* d_out, int out_size, void* d_ws, size_t ws_size,
                              hipStream_t stream) {
  (void)in_sizes; (void)n_in; (void)out_size; (void)ws_size;

  const float* x = (const float*)d_in[0];
  const float* s = (const float*)d_in[1];
  float* out = (float*)d_out;

  unsigned* flags        = (unsigned*)d_ws;
  float* thrbuf          = (float*)((char*)d_ws + WS_THR_OFF);
  unsigned char* obits   = (unsigned char*)((char*)d_ws + WS_OBIT_OFF);

  dim3 grid(NROWS), block(TPB);

  init_flags_kernel<<<1, 32, 0, stream>>>(flags);

  stats_kernel<true ><<<grid, block, 0, stream>>>(x, s, flags, thrbuf, obits, 0);
  quant_kernel<true ><<<grid, block, 0, stream>>>(x, s, flags, thrbuf, obits, out, 0);
  for (int i = 1; i < NREF; ++i) {
    stats_kernel<false><<<grid, block, 0, stream>>>(x, s, flags, thrbuf, obits, i);
    quant_kernel<false><<<grid, block, 0, stream>>>(x, s, flags, thrbuf, obits, out, i);
  }
}